// DimeNet_2439541424494
// MI455X (gfx1250) — compile-verified
//
#include <hip/hip_runtime.h>

// DimeNet Bessel radial-basis kernel for gfx1250 (MI455X), wave32.
//
// out[e][n] = env(d_e) * sin(freq[n] * d_e),  d_e = |R[i]-R[j]| / 5
// env(x)   = 1/x - 28 x^5 + 48 x^6 - 21 x^7      (envelope_exponent=5 -> p=6)
//
// Store-BW bound: 204.8 MB of output @ 23.3 TB/s ~ 8.8 us floor.
// R (1.2 MB) lives in L2/WGP$, so the random gathers are cheap.
// The d (x) freq outer product and the env row-broadcast are done with
// V_WMMA_F32_16X16X4_F32 rank-1 matmuls, which land both the product and
// the envelope in identical C/D lane layouts (lane n = col, vgpr r = row).
// Stores: wave-uniform tail check -> fast path is 16 global_store_b32 with
// immediate offsets off a single per-lane base address (no exec juggling).

typedef __attribute__((ext_vector_type(2))) float v2f;
typedef __attribute__((ext_vector_type(8))) float v8f;

__global__ __launch_bounds__(256) void dimenet_rbf_wmma(
    const float* __restrict__ R,
    const float* __restrict__ freq,
    const int*   __restrict__ idx_i,
    const int*   __restrict__ idx_j,
    float*       __restrict__ out,
    int n_edges)
{
    const int lane   = threadIdx.x & 31;              // wave32
    const int wave   = (blockIdx.x * blockDim.x + threadIdx.x) >> 5;
    const int nwaves = (gridDim.x * blockDim.x) >> 5;
    const bool lo    = lane < 16;

    // B matrices (4x16, only row K=0 nonzero). Rows stripe across lanes in
    // one VGPR; zeroing lanes 16-31 and VGPR1 kills K=1..3 under any
    // K-interleave convention.
    const float fq = freq[lane & 15];
    v2f Bf; Bf.x = lo ? fq   : 0.0f; Bf.y = 0.0f;     // B[0][n] = freq[n]
    v2f B1; B1.x = lo ? 1.0f : 0.0f; B1.y = 0.0f;     // B[0][n] = 1 (env bcast)

    const int stride = nwaves << 5;                   // edges per grid sweep
    const int n      = lane & 15;                     // output column
    const int mofs   = (lane >> 4) << 3;              // row offset: 0 or 8

    for (int base = wave << 5; base < n_edges; base += stride) {
        int e = base + lane;
        if (e >= n_edges) e = n_edges - 1;            // clamp; stores guarded in tail

        // Prefetch next chunk's index words (speculative, OOB-safe).
        __builtin_prefetch(idx_i + e + stride, 0, 0);
        __builtin_prefetch(idx_j + e + stride, 0, 0);

        const int ni = idx_i[e];
        const int nj = idx_j[e];
        const float* Ri = R + 3 * (size_t)ni;
        const float* Rj = R + 3 * (size_t)nj;
        const float dx = Ri[0] - Rj[0];
        const float dy = Ri[1] - Rj[1];
        const float dz = Ri[2] - Rj[2];

        float d2 = fmaf(dx, dx, fmaf(dy, dy, dz * dz));
        float ds = __builtin_amdgcn_sqrtf(fmaxf(d2, 0.0f)) * 0.2f;  // Dij/5

        // env(x) = 1/x + x^5 * (-28 + 48x - 21x^2)
        float x2   = ds * ds;
        float x5   = x2 * x2 * ds;
        float poly = fmaf(ds, fmaf(ds, -21.0f, 48.0f), -28.0f);
        float env  = fmaf(x5, poly, __builtin_amdgcn_rcpf(ds));

        // Move lanes 16-31's edge values down for the second 16-edge tile.
        float ds_hi  = __shfl(ds,  lane + 16, 32);    // wraps mod 32; hi lanes zeroed below
        float env_hi = __shfl(env, lane + 16, 32);

        // A matrices (16x4, only column K=0 nonzero): lane m (0-15) vgpr0 = A[m][0].
        v2f A0; A0.x = lo ? ds     : 0.0f; A0.y = 0.0f;
        v2f A1; A1.x = lo ? ds_hi  : 0.0f; A1.y = 0.0f;
        v2f E0; E0.x = lo ? env    : 0.0f; E0.y = 0.0f;
        v2f E1; E1.x = lo ? env_hi : 0.0f; E1.y = 0.0f;

        v8f z = {};
        // P[m][n] = ds_m * freq_n ; V[m][n] = env_m  (same C/D layout)
        v8f P0 = __builtin_amdgcn_wmma_f32_16x16x4_f32(false, A0, false, Bf, (short)0, z, false, false);
        v8f P1 = __builtin_amdgcn_wmma_f32_16x16x4_f32(false, A1, false, Bf, (short)0, z, false, false);
        v8f V0 = __builtin_amdgcn_wmma_f32_16x16x4_f32(false, E0, false, B1, (short)0, z, false, false);
        v8f V1 = __builtin_amdgcn_wmma_f32_16x16x4_f32(false, E1, false, B1, (short)0, z, false, false);

        // C/D layout: lane holds column n = lane&15; vgpr r holds row
        // m = r + mofs. Tile0 rows = base+m, tile1 rows = base+16+m.
        // Single per-lane base address; per-r offsets become store immediates.
        float* p = out + (((size_t)(base + mofs)) << 4) + n;

        if (base + 32 <= n_edges) {
            // Fast path (all chunks but the last): wave-uniform branch,
            // unguarded stores with immediate offsets (r*64B, +1KB for tile1).
#pragma unroll
            for (int r = 0; r < 8; ++r) {
                p[r * 16]       = V0[r] * __sinf(P0[r]);
                p[r * 16 + 256] = V1[r] * __sinf(P1[r]);
            }
        } else {
            // Tail chunk only: per-row guards.
#pragma unroll
            for (int r = 0; r < 8; ++r) {
                const float s0 = V0[r] * __sinf(P0[r]);
                const float s1 = V1[r] * __sinf(P1[r]);
                if (base + mofs + r < n_edges)      p[r * 16]       = s0;
                if (base + 16 + mofs + r < n_edges) p[r * 16 + 256] = s1;
            }
        }
    }
}

extern "C" void kernel_launch(void* const* d_in, const int* in_sizes, int n_in,
                              void* d_out, int out_size, void* d_ws, size_t ws_size,
                              hipStream_t stream) {
    const float* R     = (const float*)d_in[0];   // [N_NODES, 3] f32
    const float* freq  = (const float*)d_in[1];   // [16] f32
    const int*   idx_i = (const int*)  d_in[2];   // [E] int32 (JAX x64 off)
    const int*   idx_j = (const int*)  d_in[3];   // [E] int32
    float*       out   = (float*)d_out;           // [E, 16] f32
    const int n_edges  = in_sizes[2];

    const int threads = 256;                      // 8 wave32 waves / block
    const int edges_per_block = 256;              // 8 waves * 32 edges
    int blocks = (n_edges + edges_per_block - 1) / edges_per_block;
    if (blocks > 4096) blocks = 4096;             // ~3 grid-stride sweeps; makes
    if (blocks < 1)    blocks = 1;                // the index prefetch useful

    dimenet_rbf_wmma<<<blocks, threads, 0, stream>>>(R, freq, idx_i, idx_j, out, n_edges);
}